// ChamferDistanceL2_14293651161196
// MI455X (gfx1250) — compile-verified
//
#include <hip/hip_runtime.h>

typedef float v2f __attribute__((ext_vector_type(2)));
typedef float v8f __attribute__((ext_vector_type(8)));

#define B_  8
#define N_  4096
#define M_  4096
#define WAVES_PER_BLOCK 8
#define ROWS_PER_BLOCK  128                      // 8 waves x 16 rows
#define MCHUNK          1024                     // columns per block
#define NROWBLKS        (N_ / ROWS_PER_BLOCK)    // 32
#define NMCHUNKS        (M_ / MCHUNK)            // 4
#define INF_BITS        0x7F800000u

// ---------------------------------------------------------------------------
// Kernel 0: initialize min1bits and min2bits (contiguous) to +inf bits
// ---------------------------------------------------------------------------
__global__ __launch_bounds__(256) void chamfer_init(unsigned int* __restrict__ p) {
    int i = blockIdx.x * 256 + threadIdx.x;
    if (i < 2 * B_ * N_) p[i] = INF_BITS;
}

// ---------------------------------------------------------------------------
// Kernel 1: main tile kernel.
// Block = 8 waves = one 128-row n-slab x one 1024-col m-chunk of one batch.
// xyz2 chunk staged in LDS as float4(x,y,z,|y|^2).  Each wave: 64 m-tiles,
// one V_WMMA_F32_16X16X4_F32 per 16x16 tile (K=3 zero-padded to 4).
// ---------------------------------------------------------------------------
__global__ __launch_bounds__(256) void chamfer_main(const float* __restrict__ xyz1,
                                                    const float* __restrict__ xyz2,
                                                    unsigned int* __restrict__ min1bits,
                                                    unsigned int* __restrict__ min2bits) {
    __shared__ float4       s_pts[MCHUNK];       // 16 KB staged xyz2 chunk
    __shared__ unsigned int s_min2[MCHUNK];      //  4 KB block-local column mins

    const int id = blockIdx.x;
    const int mc = id % NMCHUNKS;
    const int nb = (id / NMCHUNKS) % NROWBLKS;
    const int b  = id / (NMCHUNKS * NROWBLKS);

    const int wave   = threadIdx.x >> 5;
    const int lane   = threadIdx.x & 31;
    const int hi     = lane >> 4;                // 0: lanes 0-15, 1: lanes 16-31
    const int r      = lane & 15;
    const int n0     = nb * ROWS_PER_BLOCK + wave * 16;
    const int m_base = mc * MCHUNK;

    // ---- stage xyz2 chunk into LDS with |y|^2 precomputed
    for (int i = threadIdx.x; i < MCHUNK; i += 256) {
        const float* p = xyz2 + ((size_t)b * M_ + m_base + i) * 3;
        const float x = p[0], y = p[1], z = p[2];
        s_pts[i]  = make_float4(x, y, z, x * x + y * y + z * z);
        s_min2[i] = INF_BITS;
    }
    __syncthreads();

    // ---- A tile (16 rows x K=4, zero-padded): lane l holds row l&15,
    //      VGPR0 = comp (hi?2:0), VGPR1 = comp (hi?3:1)
    const float* p1 = xyz1 + ((size_t)b * N_ + n0 + r) * 3;
    const float ax = p1[0], ay = p1[1], az = p1[2];
    const float x2 = ax * ax + ay * ay + az * az;
    v2f a;
    a.x = hi ? az : ax;
    a.y = hi ? 0.0f : ay;

    // |x|^2 for the 8 rows this lane's C registers cover (rows v + 8*hi)
    float x2row[8];
#pragma unroll
    for (int v = 0; v < 8; ++v) x2row[v] = __shfl(x2, v + 8 * hi, 32);

    float minrow[8];
#pragma unroll
    for (int v = 0; v < 8; ++v) minrow[v] = __builtin_inff();

    // ---- 64 m-tiles from LDS; unroll x2 for two independent WMMA chains
#pragma unroll 2
    for (int t = 0; t < MCHUNK / 16; ++t) {
        const float4 q = s_pts[(t << 4) + r];    // ds_load_b128
        v2f bv;
        bv.x = hi ? q.z : q.x;
        bv.y = hi ? 0.0f : q.y;

        v8f c = {0.f, 0.f, 0.f, 0.f, 0.f, 0.f, 0.f, 0.f};
        c = __builtin_amdgcn_wmma_f32_16x16x4_f32(
                /*neg_a=*/false, a, /*neg_b=*/false, bv,
                /*c_mod=*/(short)0, c, /*reuse_a=*/false, /*reuse_b=*/false);

        float colmin = __builtin_inff();
#pragma unroll
        for (int v = 0; v < 8; ++v) {
            float d2 = fmaxf(x2row[v] + q.w - 2.0f * c[v], 0.0f);
            minrow[v] = fminf(minrow[v], d2);    // running min over m (min1)
            colmin    = fminf(colmin, d2);       // min over this lane's 8 rows
        }
        colmin = fminf(colmin, __shfl_xor(colmin, 16, 32));
        if (!hi) atomicMin(&s_min2[(t << 4) + r], __float_as_uint(colmin));
    }

    // ---- min1: reduce across the 16 columns (xor 8..1 stays within each half)
#pragma unroll
    for (int mask = 8; mask >= 1; mask >>= 1) {
#pragma unroll
        for (int v = 0; v < 8; ++v)
            minrow[v] = fminf(minrow[v], __shfl_xor(minrow[v], mask, 32));
    }
    if (r == 0) {
#pragma unroll
        for (int v = 0; v < 8; ++v)
            atomicMin(&min1bits[(size_t)b * N_ + n0 + v + 8 * hi],
                      __float_as_uint(minrow[v]));
    }

    __syncthreads();
    // ---- flush block-local column mins: one global atomic per column per block
    for (int i = threadIdx.x; i < MCHUNK; i += 256)
        atomicMin(&min2bits[(size_t)b * M_ + m_base + i], s_min2[i]);
}

// ---------------------------------------------------------------------------
// Kernel 2: finalize.  result = (sum(min1) + sum(min2)) / (B*N)   (N == M)
// ---------------------------------------------------------------------------
__global__ __launch_bounds__(256) void chamfer_finalize(const unsigned int* __restrict__ min1bits,
                                                        const unsigned int* __restrict__ min2bits,
                                                        float* __restrict__ out) {
    __shared__ float s[256];
    const int tid = threadIdx.x;
    float sum = 0.0f;
    for (int i = tid; i < B_ * N_; i += 256) sum += __uint_as_float(min1bits[i]);
    for (int i = tid; i < B_ * M_; i += 256) sum += __uint_as_float(min2bits[i]);
    s[tid] = sum;
    __syncthreads();
    for (int stride = 128; stride > 0; stride >>= 1) {
        if (tid < stride) s[tid] += s[tid + stride];
        __syncthreads();
    }
    if (tid == 0) out[0] = s[0] / (float)(B_ * N_);
}

// ---------------------------------------------------------------------------
extern "C" void kernel_launch(void* const* d_in, const int* in_sizes, int n_in,
                              void* d_out, int out_size, void* d_ws, size_t ws_size,
                              hipStream_t stream) {
    (void)in_sizes; (void)n_in; (void)out_size; (void)ws_size;

    const float* xyz1 = (const float*)d_in[0];
    const float* xyz2 = (const float*)d_in[1];

    unsigned int* min1bits = (unsigned int*)d_ws;                  // 128 KB
    unsigned int* min2bits = min1bits + (size_t)B_ * N_;           // 128 KB

    chamfer_init<<<(2 * B_ * N_ + 255) / 256, 256, 0, stream>>>(min1bits);
    chamfer_main<<<B_ * NROWBLKS * NMCHUNKS, 256, 0, stream>>>(xyz1, xyz2, min1bits, min2bits);
    chamfer_finalize<<<1, 256, 0, stream>>>(min1bits, min2bits, (float*)d_out);
}